// ResonantMemoryField_80728205296349
// MI455X (gfx1250) — compile-verified
//
#include <hip/hip_runtime.h>

// Resonant Memory Field scan for MI455X (gfx1250).
// Layout: 128 blocks x 32 threads (1 wave/block). Block = (batch b, d-group).
// Each thread owns one (b, d) phasor chain of length T=4096.
// - Index row staged to LDS via gfx1250 async global->LDS DMA (ASYNCcnt).
// - Codebook gathers double-buffered in registers (latency off scan chain).
// - Normalization: single v_rsq_f32 + v_min on the critical path.
// - History written with non-temporal stores (write-once, keep L2 for table).

typedef __attribute__((ext_vector_type(2))) float v2f;

#define RMF_B       32
#define RMF_T       4096
#define RMF_D       128
#define RMF_BUCKETS 65536
#define RMF_U       8          // steps per prefetch buffer
#define ETA_F       0.1f
#define OME_F       0.9f       // 1 - eta

__global__ __launch_bounds__(32)
void rmf_scan_kernel(const float* __restrict__ code,   // (65536, 128, 2) f32
                     const int*   __restrict__ idx,    // (32, 4096) i32
                     float*       __restrict__ out)    // (32, 4096, 128, 2) f32
{
    __shared__ int s_idx[RMF_T];

    const int lane = threadIdx.x;          // 0..31
    const int b    = blockIdx.x >> 2;      // batch 0..31
    const int dgrp = blockIdx.x & 3;       // d-group 0..3
    const int d    = (dgrp << 5) + lane;   // 0..127

    const int* idx_row = idx + b * RMF_T;
    const unsigned lds_base = (unsigned)(size_t)(&s_idx[0]); // low 32b = LDS offset

    // ---- Stage this batch's 4096 indices into LDS: async DMA (gfx1250) ----
    // 32 x b128 per lane-group: 4096 ints = 16KB. Tracked by ASYNCcnt.
#pragma unroll
    for (int i = 0; i < RMF_T / (32 * 4); ++i) {
        unsigned goff  = (unsigned)((((i << 5) + lane)) << 4);  // *16 bytes
        unsigned laddr = lds_base + goff;
        asm volatile("global_load_async_to_lds_b128 %0, %1, %2"
                     :: "v"(laddr), "v"(goff), "s"(idx_row) : "memory");
    }
    asm volatile("s_wait_asynccnt 0" ::: "memory");
    __syncthreads();   // single-wave WG: lowers to nop, kept for clarity

    const float* codeD = code + (d << 1);                       // column base
    float* outp = out + ((size_t)b * RMF_T * RMF_D + d) * 2;    // (b, 0, d, 0)

    float fr = 1.0f, fi = 0.0f;
    v2f bufA[RMF_U], bufB[RMF_U];

#define RMF_LOAD_CHUNK(T0, DST)                                              \
    {                                                                        \
        _Pragma("unroll")                                                    \
        for (int u = 0; u < RMF_U; ++u) {                                    \
            int ii = s_idx[(T0) + u];                                        \
            ii = ii < 0 ? 0 : ii;                                            \
            ii = ii > (RMF_BUCKETS - 1) ? (RMF_BUCKETS - 1) : ii;            \
            (DST)[u] = *(const v2f*)(codeD + ((size_t)ii << 8));             \
        }                                                                    \
    }

#define RMF_STEP(T0, SRC)                                                    \
    {                                                                        \
        _Pragma("unroll")                                                    \
        for (int u = 0; u < RMF_U; ++u) {                                    \
            v2f drv = (SRC)[u];                                              \
            float nr = fmaf(OME_F, fr, ETA_F * drv.x);                       \
            float ni = fmaf(OME_F, fi, ETA_F * drv.y);                       \
            float m2 = fmaf(nr, nr, ni * ni);                                \
            float inv = fminf(__builtin_amdgcn_rsqf(m2), 1.0e6f);            \
            v2f prev; prev.x = fr; prev.y = fi;                              \
            __builtin_nontemporal_store(prev,                                \
                (v2f*)(outp + (size_t)((T0) + u) * (2 * RMF_D)));            \
            fr = nr * inv;                                                   \
            fi = ni * inv;                                                   \
        }                                                                    \
    }

    RMF_LOAD_CHUNK(0, bufA);

    for (int t0 = 0; t0 < RMF_T; t0 += 2 * RMF_U) {
        // L2 warm hint ~4 chunks ahead (global_prefetch_b8), 16 rows x 512B
        {
            int tp = t0 + 4 * RMF_U + (lane >> 1);
            if (tp < RMF_T) {
                int ii = s_idx[tp];
                ii = ii < 0 ? 0 : ii;
                ii = ii > (RMF_BUCKETS - 1) ? (RMF_BUCKETS - 1) : ii;
                const char* p = (const char*)code + ((size_t)ii << 10)
                                + ((size_t)(lane & 1) << 9);
                __builtin_prefetch(p, 0, 0);
            }
        }
        RMF_LOAD_CHUNK(t0 + RMF_U, bufB);          // fetch next while computing
        RMF_STEP(t0, bufA);
        if (t0 + 2 * RMF_U < RMF_T) RMF_LOAD_CHUNK(t0 + 2 * RMF_U, bufA);
        RMF_STEP(t0 + RMF_U, bufB);
    }
}

extern "C" void kernel_launch(void* const* d_in, const int* in_sizes, int n_in,
                              void* d_out, int out_size, void* d_ws, size_t ws_size,
                              hipStream_t stream)
{
    (void)in_sizes; (void)n_in; (void)out_size; (void)d_ws; (void)ws_size;
    const float* code = (const float*)d_in[0];   // bucket_code f32
    const int*   idx  = (const int*)d_in[1];     // idx_r (int32 per harness rule)
    float*       out  = (float*)d_out;           // (B,T,128,2) f32

    dim3 grid(RMF_B * (RMF_D / 32));   // 128 blocks
    dim3 block(32);                    // 1 wave per block (wave32)
    rmf_scan_kernel<<<grid, block, 0, stream>>>(code, idx, out);
}